// SpeakerEncoder_78151224918456
// MI455X (gfx1250) — compile-verified
//
#include <hip/hip_runtime.h>
#include <cstdint>

// ---------------------------------------------------------------------------
// SpeakerEncoder on gfx1250 (CDNA5, wave32, WMMA, TDM, async-LDS)
// dims
#define Bd    8
#define TMEL  400
#define NMEL  80
#define LTXT  128
#define DTXT  512
#define Ed    256
#define HIDd  256
#define Hh    128
#define EMBd  128
#define NSPK  100

typedef __attribute__((ext_vector_type(2))) float v2f;
typedef __attribute__((ext_vector_type(8))) float v8f;
typedef __attribute__((ext_vector_type(4))) unsigned int u32x4;
typedef __attribute__((ext_vector_type(4))) int i32x4;
typedef __attribute__((ext_vector_type(8))) int i32x8;

// async global->LDS builtins take int pointers in AS1 (global) / AS3 (LDS)
typedef __attribute__((address_space(1))) int gas_i32;
typedef __attribute__((address_space(3))) int las_i32;

#if defined(__has_builtin)
#if __has_builtin(__builtin_amdgcn_tensor_load_to_lds)
#define HAVE_TDM 1
#endif
#if __has_builtin(__builtin_amdgcn_global_load_async_to_lds_b128)
#define HAVE_ASYNC_LDS 1
#endif
#if __has_builtin(__builtin_amdgcn_global_load_async_to_lds_b32)
#define HAVE_ASYNC_B32 1
#endif
#endif

// report chosen data-mover paths in stderr (host pass reports fallback; the
// device (gfx1250) pass report is the authoritative one)
#if defined(HAVE_TDM)
#warning "CDNA5 device path: tensor_load_to_lds (TDM) enabled"
#elif defined(HAVE_ASYNC_LDS)
#warning "CDNA5 device path: global_load_async_to_lds_b128 enabled (no TDM builtin)"
#else
#warning "CDNA5 device path: scalar LDS copy fallback (no TDM/async builtins)"
#endif
#if defined(HAVE_ASYNC_B32)
#warning "CDNA5 device path: global_load_async_to_lds_b32 enabled"
#endif

__device__ __forceinline__ v8f wmma4(v2f a, v2f b, v8f c) {
  // D = A(16x4,f32) * B(4x16,f32) + C(16x16,f32)
  return __builtin_amdgcn_wmma_f32_16x16x4_f32(false, a, false, b, (short)0, c,
                                               false, false);
}

__device__ __forceinline__ float sigf(float x) { return 1.f / (1.f + __expf(-x)); }

__device__ __forceinline__ void wait_asynccnt0() {
#if defined(__has_builtin) && __has_builtin(__builtin_amdgcn_s_wait_asynccnt)
  __builtin_amdgcn_s_wait_asynccnt(0);
#else
  asm volatile("s_wait_asynccnt 0x0" ::: "memory");
#endif
}

// ---------------------------------------------------------------------------
// Generic strided/batched WMMA f32 GEMM: C[b] = A[b](MxK) * B[b](KxN) + bias
// A[m,k] = A[b*aBS + m*aRS + k*aCS]; B[k,n] = B[b*bBS + k*bRS + n*bCS]
// block = 128 threads = 4 waves, each wave owns one 16x16 tile.
// grid = (N/64, ceil(M/16), batch). K must be a multiple of 8.
__global__ void gemm_wmma_f32(const float* __restrict__ A, long aRS, long aCS, long aBS,
                              const float* __restrict__ Bm, long bRS, long bCS, long bBS,
                              const float* __restrict__ bias, long biasBS,
                              float* __restrict__ C, long cRS, long cBS,
                              int M, int N, int K) {
  const int wid  = threadIdx.x >> 5;
  const int lane = threadIdx.x & 31;
  const int tn   = blockIdx.x * 4 + wid;
  const int n0   = tn * 16;
  const int m0   = blockIdx.y * 16;
  const int bz   = blockIdx.z;
  if (n0 >= N) return;                     // wave-uniform
  const float* Ab = A + (long)bz * aBS;
  const float* Bb = Bm + (long)bz * bBS;
  const int lm = lane & 15;
  const int hi = lane >> 4;
  const int mrow  = m0 + lm;
  const int ncol  = n0 + lm;
  const bool mval = (mrow < M);
  const int mr    = mval ? mrow : (M - 1);  // clamp, zero via select
  v8f acc = {0.f, 0.f, 0.f, 0.f, 0.f, 0.f, 0.f, 0.f};
  for (int k0 = 0; k0 < K; k0 += 8) {
    const int ka0 = k0 + 2 * hi;
    const int ka1 = k0 + 4 + 2 * hi;
    v2f a0, b0, a1, b1;
    float a0x = Ab[(long)mr * aRS + (long)ka0 * aCS];
    float a0y = Ab[(long)mr * aRS + (long)(ka0 + 1) * aCS];
    float a1x = Ab[(long)mr * aRS + (long)ka1 * aCS];
    float a1y = Ab[(long)mr * aRS + (long)(ka1 + 1) * aCS];
    b0.x = Bb[(long)ka0 * bRS + (long)ncol * bCS];
    b0.y = Bb[(long)(ka0 + 1) * bRS + (long)ncol * bCS];
    b1.x = Bb[(long)ka1 * bRS + (long)ncol * bCS];
    b1.y = Bb[(long)(ka1 + 1) * bRS + (long)ncol * bCS];
    a0.x = mval ? a0x : 0.f;
    a0.y = mval ? a0y : 0.f;
    a1.x = mval ? a1x : 0.f;
    a1.y = mval ? a1y : 0.f;
    if (k0 + 8 < K) {                      // gfx1250 global_prefetch_b8
      __builtin_prefetch(&Bb[(long)(ka0 + 8) * bRS + (long)ncol * bCS], 0, 1);
    }
    acc = wmma4(a0, b0, acc);
    acc = wmma4(a1, b1, acc);
  }
  const float bb = bias ? bias[(long)bz * biasBS + ncol] : 0.f;
  float* Cb = C + (long)bz * cBS;
  const int mbase = m0 + 8 * hi;
#pragma unroll
  for (int vi = 0; vi < 8; ++vi) {
    const int mo = mbase + vi;
    if (mo < M) Cb[(long)mo * cRS + ncol] = acc[vi] + bb;
  }
}

// ---------------------------------------------------------------------------
// prep: w = Wout^T v ; c = bout.v ; bsum[l,d] = bih+bhh   (1 block, 256 thr)
__global__ void prep_kernel(const float* __restrict__ Wout, const float* __restrict__ bout,
                            const float* __restrict__ v, const float* __restrict__ bih,
                            const float* __restrict__ bhh, float* __restrict__ wv,
                            float* __restrict__ cout, float* __restrict__ bsum) {
  const int tid = threadIdx.x;
  float a = 0.f;
  for (int i = 0; i < Ed; ++i) a += Wout[i * Ed + tid] * v[i];
  wv[tid] = a;
  if (tid == 0) {
    float c = 0.f;
    for (int i = 0; i < Ed; ++i) c += bout[i] * v[i];
    cout[0] = c;
  }
  for (int i = tid; i < 4 * 4 * Hh; i += 256) bsum[i] = bih[i] + bhh[i];
}

// ---------------------------------------------------------------------------
// attention scores + softmax: one block per (b,l); 256 thr = 8 waves.
// scores[b,l,t] = softmax_t( mask(sum_e tanh(x1[b,t,e]+q1[b,l,e])*w[e] + c) )
__global__ void attn_scores(const float* __restrict__ x1, const float* __restrict__ q1,
                            const float* __restrict__ wv, const float* __restrict__ cptr,
                            const int* __restrict__ ilen, float* __restrict__ scores) {
  __shared__ float qrow[Ed];
  __shared__ float wsh[Ed];
  __shared__ float srow[TMEL];
  __shared__ float red[8];
  const int bl = blockIdx.x;           // 0..B*L-1
  const int b = bl >> 7;
  const int tid = threadIdx.x;
  const int wid = tid >> 5, lane = tid & 31;
#if defined(HAVE_ASYNC_B32)
  {  // stage q1 row and w via CDNA5 async global->LDS (ASYNCcnt)
    __builtin_amdgcn_global_load_async_to_lds_b32(
        (gas_i32*)(uintptr_t)(q1 + (long)bl * Ed + tid),
        (las_i32*)(unsigned int)(uintptr_t)&qrow[tid], 0, 0);
    __builtin_amdgcn_global_load_async_to_lds_b32(
        (gas_i32*)(uintptr_t)(wv + tid),
        (las_i32*)(unsigned int)(uintptr_t)&wsh[tid], 0, 0);
    wait_asynccnt0();
  }
#else
  qrow[tid] = q1[(long)bl * Ed + tid];
  wsh[tid] = wv[tid];
#endif
  __syncthreads();
  const float cadd = cptr[0];
  const int len = ilen[b];
  const float4 q0 = *(const float4*)&qrow[lane * 8];
  const float4 q1v = *(const float4*)&qrow[lane * 8 + 4];
  const float4 w0 = *(const float4*)&wsh[lane * 8];
  const float4 w1 = *(const float4*)&wsh[lane * 8 + 4];
  const float* xb = x1 + (long)b * TMEL * Ed;
  for (int t = wid; t < TMEL; t += 8) {
    const float* xr = xb + (long)t * Ed + lane * 8;
    const float4 xa = *(const float4*)xr;
    const float4 xc = *(const float4*)(xr + 4);
    float s = tanhf(xa.x + q0.x) * w0.x + tanhf(xa.y + q0.y) * w0.y +
              tanhf(xa.z + q0.z) * w0.z + tanhf(xa.w + q0.w) * w0.w +
              tanhf(xc.x + q1v.x) * w1.x + tanhf(xc.y + q1v.y) * w1.y +
              tanhf(xc.z + q1v.z) * w1.z + tanhf(xc.w + q1v.w) * w1.w;
    for (int off = 16; off; off >>= 1) s += __shfl_xor(s, off, 32);
    if (lane == 0) srow[t] = (t < len) ? (s + cadd) : -__builtin_inff();
  }
  __syncthreads();
  // block softmax over TMEL
  float m = -__builtin_inff();
  for (int t = tid; t < TMEL; t += 256) m = fmaxf(m, srow[t]);
  for (int off = 16; off; off >>= 1) m = fmaxf(m, __shfl_xor(m, off, 32));
  if (lane == 0) red[wid] = m;
  __syncthreads();
  float gmax = red[0];
#pragma unroll
  for (int i = 1; i < 8; ++i) gmax = fmaxf(gmax, red[i]);
  __syncthreads();
  float ls = 0.f;
  for (int t = tid; t < TMEL; t += 256) {
    const float e = __expf(srow[t] - gmax);
    srow[t] = e;
    ls += e;
  }
  for (int off = 16; off; off >>= 1) ls += __shfl_xor(ls, off, 32);
  if (lane == 0) red[wid] = ls;
  __syncthreads();
  float tot = red[0];
#pragma unroll
  for (int i = 1; i < 8; ++i) tot += red[i];
  const float inv = 1.f / tot;
  float* so = scores + (long)bl * TMEL;
  for (int t = tid; t < TMEL; t += 256) so[t] = srow[t] * inv;
}

// ---------------------------------------------------------------------------
// elementwise add: h0 = q1 + ctx
__global__ void add2_kernel(const float* __restrict__ a, const float* __restrict__ b,
                            float* __restrict__ o, int n) {
  const int i = blockIdx.x * 256 + threadIdx.x;
  if (i < n) o[i] = a[i] + b[i];
}

// ---------------------------------------------------------------------------
// LSTM scan, one direction per block (grid.x = 2), 512 threads = 16 waves.
// Whh (512x128 f32 = 256KB) is DMA'd by the Tensor Data Mover into the 320KB
// CDNA5 LDS (fallback: async global->LDS copies, then plain loads). Per step
// the recurrent GEMM h@Whh^T (M=16 pad, N=512, K=128) runs on WMMA from LDS.
// Gate order i,f,g,o (PyTorch). Bwd direction uses pack_padded reversal:
// input row = clamp(len-1-s, 0), output scattered back with the same map.
__global__ void lstm_scan(const float* __restrict__ Gpre, long gDirStride,
                          const float* __restrict__ whh, const int* __restrict__ tlen,
                          float* __restrict__ out /* [B, L, 2H] */) {
  extern __shared__ float smem[];
  float* whhS = smem;                         // 512*128
  float* hS = smem + 512 * Hh;                // 16*128 (rows 8..15 stay zero)
  float* cS = hS + 16 * Hh;                   // 8*128
  float* gS = cS + 8 * Hh;                    // 8*512
  const int dir = blockIdx.x;
  const int tid = threadIdx.x;                // 512
  const float* W = whh + (long)dir * (4 * Hh * Hh);

#if defined(HAVE_TDM)
  // ---- Tensor Data Mover: 2D tile 128x512 f32, contiguous, -> LDS ----
  if (tid < 32) {  // wave 0 issues; EXEC ignored by TDM, TENSORcnt per-wave
    const unsigned long long ga = (unsigned long long)(uintptr_t)W;
    const unsigned int ldsb = (unsigned int)(uintptr_t)whhS;  // LDS byte offset
    u32x4 g0;
    g0.x = 1u;                                         // count=1 (valid user D#)
    g0.y = ldsb;                                       // lds_addr
    g0.z = (unsigned int)(ga & 0xFFFFFFFFull);         // global_addr[31:0]
    g0.w = (unsigned int)((ga >> 32) & 0x1FFFFFFull)   // global_addr[56:32]
           | (2u << 30);                               // type = 2 ("image")
    i32x8 g1;
    g1[0] = (int)(2u << 16);          // wg_mask=0, data_size=2 (4 bytes)
    g1[1] = (int)((Hh & 0xFFFFu) << 16);         // tensor_dim0[15:0] @ bits63:48
    g1[2] = (int)((Hh >> 16) | ((512u & 0xFFFFu) << 16)); // dim0 hi | dim1 lo
    g1[3] = (int)((512u >> 16) | ((unsigned)Hh << 16));   // dim1 hi | tile_dim0
    g1[4] = (int)512;                 // tile_dim1=512, tile_dim2=0
    g1[5] = (int)Hh;                  // tensor_dim0_stride[31:0] = 128
    g1[6] = 0;                        // stride hi | dim1_stride lo (unused, 2D)
    g1[7] = 0;
    i32x4 gz = {0, 0, 0, 0};
#if __clang_major__ >= 23
    i32x8 gz8 = {0, 0, 0, 0, 0, 0, 0, 0};
    __builtin_amdgcn_tensor_load_to_lds(g0, g1, gz, gz, gz8, 0);
#else
    __builtin_amdgcn_tensor_load_to_lds(g0, g1, gz, gz, 0);
#endif
    __builtin_amdgcn_s_wait_tensorcnt(0);
  }
#elif defined(HAVE_ASYNC_LDS)
  {
    for (int i = tid * 4; i < 4 * Hh * Hh; i += 512 * 4) {
      __builtin_amdgcn_global_load_async_to_lds_b128(
          (gas_i32*)(uintptr_t)(W + i),
          (las_i32*)(unsigned int)(uintptr_t)(whhS + i), 0, 0);
    }
    wait_asynccnt0();
  }
#else
  for (int i = tid; i < 4 * Hh * Hh; i += 512) whhS[i] = W[i];
#endif

  for (int i = tid; i < 16 * Hh; i += 512) hS[i] = 0.f;
  for (int i = tid; i < 8 * Hh; i += 512) cS[i] = 0.f;
  __syncthreads();

  const int wid = tid >> 5, lane = tid & 31;
  const int lm = lane & 15, hi = lane >> 4;
  const float* Gd = Gpre + (long)dir * gDirStride;
  for (int s = 0; s < LTXT; ++s) {
    // ---- gate_rec = h @ Whh^T via WMMA (all lanes active at wmma) ----
    for (int t2 = 0; t2 < 2; ++t2) {
      const int n0 = (wid * 2 + t2) * 16;
      v8f acc = {0.f, 0.f, 0.f, 0.f, 0.f, 0.f, 0.f, 0.f};
      for (int kk = 0; kk < Hh / 8; ++kk) {
        const int ka0 = 8 * kk + 2 * hi;
        const int ka1 = 8 * kk + 4 + 2 * hi;
        const v2f a0 = *(const v2f*)&hS[lm * Hh + ka0];
        const v2f b0 = *(const v2f*)&whhS[(n0 + lm) * Hh + ka0];  // B[k,n]=Whh[n,k]
        const v2f a1 = *(const v2f*)&hS[lm * Hh + ka1];
        const v2f b1 = *(const v2f*)&whhS[(n0 + lm) * Hh + ka1];
        acc = wmma4(a0, b0, acc);
        acc = wmma4(a1, b1, acc);
      }
      if (lane < 16) {  // rows m=0..7 live in lanes 0..15 / regs 0..7
#pragma unroll
        for (int vi = 0; vi < 8; ++vi) gS[vi * 512 + n0 + lm] = acc[vi];
      }
    }
    __syncthreads();
    // ---- pointwise LSTM cell update ----
    for (int r = 0; r < 2; ++r) {
      const int idx = tid + r * 512;  // 0..1023 -> (b,e)
      const int b = idx >> 7, e = idx & 127;
      const int len = tlen[b];
      int trow;
      if (dir == 0) trow = s;
      else { const int t = len - 1 - s; trow = t < 0 ? 0 : t; }
      const float* g = Gd + ((long)(b * LTXT + trow)) * (4 * Hh);
      const float gi = gS[b * 512 + e]            + g[e];
      const float gf = gS[b * 512 + Hh + e]       + g[Hh + e];
      const float gg = gS[b * 512 + 2 * Hh + e]   + g[2 * Hh + e];
      const float go = gS[b * 512 + 3 * Hh + e]   + g[3 * Hh + e];
      const float cc = sigf(gf) * cS[b * Hh + e] + sigf(gi) * tanhf(gg);
      const float hh = sigf(go) * tanhf(cc);
      cS[b * Hh + e] = cc;
      hS[b * Hh + e] = hh;
      if (dir == 0) {
        out[((long)(b * LTXT + s)) * HIDd + e] = hh;
      } else {
        if (s < len) out[((long)(b * LTXT + (len - 1 - s))) * HIDd + Hh + e] = hh;
        if (s == 0) {
          for (int j = len; j < LTXT; ++j)
            out[((long)(b * LTXT + j)) * HIDd + Hh + e] = hh;
        }
      }
    }
    __syncthreads();
  }
}

// ---------------------------------------------------------------------------
// masked mean pool -> tanh proj -> (logits, L2-normalized embedding)
__global__ void pool_head(const float* __restrict__ hfin, const int* __restrict__ tlen,
                          const float* __restrict__ p1w, const float* __restrict__ p1b,
                          const float* __restrict__ p2w, const float* __restrict__ p2b,
                          float* __restrict__ logits, float* __restrict__ emb) {
  __shared__ float pooled[HIDd];
  __shared__ float outv[EMBd];
  __shared__ float nrm;
  const int b = blockIdx.x, tid = threadIdx.x;  // 256 threads
  const int len = tlen[b];
  const float* hb = hfin + (long)b * LTXT * HIDd;
  float s = 0.f;
  for (int j = 0; j < len; ++j) s += hb[(long)j * HIDd + tid];
  pooled[tid] = s / (float)len;
  __syncthreads();
  if (tid < EMBd) {
    float a = p1b[tid];
    for (int k = 0; k < HIDd; ++k) a += pooled[k] * p1w[tid * HIDd + k];
    outv[tid] = tanhf(a);
  }
  __syncthreads();
  if (tid == 0) {
    float ss = 0.f;
    for (int i = 0; i < EMBd; ++i) ss += outv[i] * outv[i];
    nrm = 1.f / sqrtf(ss);
  }
  __syncthreads();
  if (tid < EMBd) emb[b * EMBd + tid] = outv[tid] * nrm;
  if (tid < NSPK) {
    float a = p2b[tid];
    for (int k = 0; k < EMBd; ++k) a += outv[k] * p2w[tid * EMBd + k];
    logits[b * NSPK + tid] = a;
  }
}

// ---------------------------------------------------------------------------
extern "C" void kernel_launch(void* const* d_in, const int* in_sizes, int n_in,
                              void* d_out, int out_size, void* d_ws, size_t ws_size,
                              hipStream_t stream) {
  const float* x       = (const float*)d_in[0];   // [8,80,400]
  const int*   ilen    = (const int*)d_in[1];     // [8]
  const float* text    = (const float*)d_in[2];   // [8,128,512]
  const int*   tlen    = (const int*)d_in[3];     // [8]
  const float* Wx      = (const float*)d_in[4];   // [256,80]
  const float* bx      = (const float*)d_in[5];
  const float* Wq      = (const float*)d_in[6];   // [256,512]
  const float* bq      = (const float*)d_in[7];
  const float* Wout    = (const float*)d_in[8];   // [256,256]
  const float* bout    = (const float*)d_in[9];
  const float* vv      = (const float*)d_in[10];  // [256,1]
  const float* wih     = (const float*)d_in[11];  // [2,2,512,256]
  const float* whh     = (const float*)d_in[12];  // [2,2,512,128]
  const float* bih     = (const float*)d_in[13];  // [2,2,512]
  const float* bhh     = (const float*)d_in[14];  // [2,2,512]
  const float* p1w     = (const float*)d_in[15];  // [128,256]
  const float* p1b     = (const float*)d_in[16];
  const float* p2w     = (const float*)d_in[17];  // [100,128]
  const float* p2b     = (const float*)d_in[18];
  (void)in_sizes; (void)n_in; (void)out_size; (void)ws_size;

  float* outF   = (float*)d_out;
  float* logits = outF;                         // [8,100]
  float* emb    = outF + Bd * NSPK;             // [8,128]
  float* scores = outF + Bd * NSPK + Bd * EMBd; // [8,128,400]

  // workspace layout (floats)
  float* ws   = (float*)d_ws;
  float* wv   = ws;                 // 256
  float* csc  = ws + 256;           // 1
  float* bsum = ws + 512;           // 2048
  float* x1   = ws + 4096;          // 819200
  float* q1   = x1 + Bd * TMEL * Ed;     // 262144
  float* hA   = q1 + Bd * LTXT * Ed;     // 262144
  float* hB   = hA + Bd * LTXT * HIDd;   // 262144
  float* ctx  = hB + Bd * LTXT * HIDd;   // 262144
  float* Gpre = ctx + Bd * LTXT * Ed;    // 2*1024*512

  const int SCAN_LDS = (512 * Hh + 16 * Hh + 8 * Hh + 8 * 512) * 4;  // 290,816 B
  (void)hipFuncSetAttribute((const void*)lstm_scan,
                            hipFuncAttributeMaxDynamicSharedMemorySize, SCAN_LDS);

  // 0) small precompute
  prep_kernel<<<1, 256, 0, stream>>>(Wout, bout, vv, bih, bhh, wv, csc, bsum);

  // 1) x1 = mels @ Wx^T + bx : A[t,m]=x[b,m,t] (aRS=1,aCS=400); B[k,n]=Wx[n,k]
  gemm_wmma_f32<<<dim3(Ed / 64, TMEL / 16, Bd), 128, 0, stream>>>(
      x, 1, TMEL, (long)NMEL * TMEL, Wx, 1, NMEL, 0, bx, 0,
      x1, Ed, (long)TMEL * Ed, TMEL, Ed, NMEL);

  // 2) q1 = text @ Wq^T + bq : [1024,512]x[512,256]
  gemm_wmma_f32<<<dim3(Ed / 64, (Bd * LTXT) / 16, 1), 128, 0, stream>>>(
      text, DTXT, 1, 0, Wq, 1, DTXT, 0, bq, 0,
      q1, Ed, 0, Bd * LTXT, Ed, DTXT);

  // 3) scores (tanh-additive attention + masked softmax) -> d_out
  attn_scores<<<Bd * LTXT, 256, 0, stream>>>(x1, q1, wv, csc, ilen, scores);

  // 4) contexts = scores @ x1 (batched) : [128,400]x[400,256]
  gemm_wmma_f32<<<dim3(Ed / 64, LTXT / 16, Bd), 128, 0, stream>>>(
      scores, TMEL, 1, (long)LTXT * TMEL, x1, Ed, 1, (long)TMEL * Ed, nullptr, 0,
      ctx, Ed, (long)LTXT * Ed, LTXT, Ed, TMEL);

  // 5) h0 = q1 + contexts
  add2_kernel<<<(Bd * LTXT * Ed) / 256, 256, 0, stream>>>(q1, ctx, hA, Bd * LTXT * Ed);

  // 6/7) two BiLSTM layers: hoisted input GEMM (both dirs batched) + LDS scan
  const float* layerIn[2] = {hA, hB};
  float* layerOut[2] = {hB, hA};
  for (int l = 0; l < 2; ++l) {
    gemm_wmma_f32<<<dim3((4 * Hh) / 64, (Bd * LTXT) / 16, 2), 128, 0, stream>>>(
        layerIn[l], HIDd, 1, 0,
        wih + (long)l * 2 * 4 * Hh * HIDd, 1, HIDd, (long)4 * Hh * HIDd,
        bsum + (long)l * 2 * 4 * Hh, 4 * Hh,
        Gpre, 4 * Hh, (long)Bd * LTXT * 4 * Hh,
        Bd * LTXT, 4 * Hh, HIDd);
    lstm_scan<<<2, 512, SCAN_LDS, stream>>>(
        Gpre, (long)Bd * LTXT * 4 * Hh,
        whh + (long)l * 2 * 4 * Hh * Hh, tlen, layerOut[l]);
  }

  // 8) masked mean pool + heads -> logits, embeddings
  pool_head<<<Bd, 256, 0, stream>>>(hA, tlen, p1w, p1b, p2w, p2b, logits, emb);
}